// SpodNet_61692910239921
// MI455X (gfx1250) — compile-verified
//
#include <hip/hip_runtime.h>
#include <stdint.h>

// ---------------------------------------------------------------------------
// SpodNet one-pass for B=262144 batched 10x10 SPD matrices.
//   Phase A (per lane): Cholesky -> trtri (th -> mi), lauum streamed straight
//     into the lane's LDS slot (packed 55 floats)  -> no W register array.
//   Phase B (per lane): reload packed W from LDS, 10 rank-1 column passes.
//   The LDS round-trip physically decouples the two phases' register webs.
// Data movement: per-wave TDM (tensor_load_to_lds / tensor_store_from_lds)
//   for the 12.8KB wave slices, synced with s_wait_tensorcnt; noise staged in
//   LDS; launch_bounds(128,1); v_rcp/v_rsq + NR for div/sqrt.
// ---------------------------------------------------------------------------

#define BLOCK_T 128              // 4 waves of 32; 128 matrices per block
#define MAT_F   100              // floats per 10x10 matrix
#define DW_PER_WAVE (32 * MAT_F) // 3200 dwords per wave slice
#define GY 1.0f

#if defined(__HIP_DEVICE_COMPILE__) && defined(__gfx1250__) && \
    __has_builtin(__builtin_amdgcn_tensor_load_to_lds)
#define HAVE_TDM 1
#else
#define HAVE_TDM 0
#endif

#if HAVE_TDM
typedef __attribute__((ext_vector_type(4))) unsigned int tdm_u32x4;
typedef __attribute__((ext_vector_type(8))) int          tdm_i32x8;
typedef __attribute__((ext_vector_type(4))) int          tdm_i32x4;

// Build a 1D TDM descriptor (D#) per CDNA5 ISA:
//  group0: count=1 | lds_addr | global_addr[31:0] | global_addr[56:32], type=2
//  group1: data_size=4B; tensor_dim0 = tile_dim0 = ndw; tensor_dim1 = 1;
//          tile_dim1/2 unused; dim0 stride = ndw
//  group2/3: benign (dims beyond tile unused)
__device__ __forceinline__ void tdm_copy_1d(bool to_lds, unsigned lds_byte,
                                            unsigned long long gaddr,
                                            unsigned ndw) {
  tdm_u32x4 g0 = {
      1u,                                   // count=1 (valid), no gather
      lds_byte,                             // LDS byte address
      (unsigned)gaddr,                      // global addr [31:0]
      (unsigned)(((gaddr >> 32) & 0x01FFFFFFull) | (2u << 30)) // [56:32]|type=2
  };
  tdm_i32x8 g1 = {
      (int)(2u << 16),                      // data_size = 4 bytes; mask=0
      (int)((ndw & 0xFFFFu) << 16),         // tensor_dim0[15:0] @ bits[63:48]
      (int)(((ndw >> 16) & 0xFFFFu) | (1u << 16)), // dim0 hi16 | tensor_dim1=1
      (int)((ndw & 0xFFFFu) << 16),         // tensor_dim1 hi16=0 | tile_dim0
      0,                                    // tile_dim1=0, tile_dim2=0 (unused)
      (int)ndw,                             // tensor_dim0_stride lo32
      0, 0                                  // stride hi / dim1 stride
  };
  tdm_i32x4 g2 = {1, 1, 0, 0};              // tensor_dim2/3 benign, tile_dim3=0
  tdm_i32x4 g3 = {0, 0, 0, 0};
#if __clang_major__ >= 23
  tdm_i32x8 gz = {0, 0, 0, 0, 0, 0, 0, 0};
  if (to_lds) __builtin_amdgcn_tensor_load_to_lds(g0, g1, g2, g3, gz, 0);
  else        __builtin_amdgcn_tensor_store_from_lds(g0, g1, g2, g3, gz, 0);
#else
  if (to_lds) __builtin_amdgcn_tensor_load_to_lds(g0, g1, g2, g3, 0);
  else        __builtin_amdgcn_tensor_store_from_lds(g0, g1, g2, g3, 0);
#endif
}
#endif // HAVE_TDM

__device__ __forceinline__ void wait_tensorcnt0() {
#if defined(__HIP_DEVICE_COMPILE__) && defined(__gfx1250__)
#if __has_builtin(__builtin_amdgcn_s_wait_tensorcnt)
  __builtin_amdgcn_s_wait_tensorcnt(0);
#else
  asm volatile("s_wait_tensorcnt 0x0" ::: "memory");
#endif
#endif
}

// Scheduler fence at major phase boundaries only.
__device__ __forceinline__ void sched_fence() {
#if defined(__HIP_DEVICE_COMPILE__) && \
    __has_builtin(__builtin_amdgcn_sched_barrier)
  __builtin_amdgcn_sched_barrier(0);
#endif
}

// v_rcp_f32 + one Newton step (~0.5 ulp) instead of the ~9-op IEEE div chain.
__device__ __forceinline__ float fast_rcp(float x) {
#if defined(__HIP_DEVICE_COMPILE__) && __has_builtin(__builtin_amdgcn_rcpf)
  float r = __builtin_amdgcn_rcpf(x);
  return r * (2.0f - x * r);
#else
  return 1.0f / x;
#endif
}
// v_rsq_f32 + one Newton step instead of guarded sqrt + div.
__device__ __forceinline__ float fast_rsqrt(float x) {
#if defined(__HIP_DEVICE_COMPILE__) && __has_builtin(__builtin_amdgcn_rsqf)
  float r = __builtin_amdgcn_rsqf(x);
  return r * (1.5f - 0.5f * x * r * r);
#else
  return 1.0f / sqrtf(x);
#endif
}

// Packed symmetric lower-triangular index, i >= j.
__host__ __device__ __forceinline__ constexpr int IDX(int i, int j) {
  return i * (i + 1) / 2 + j;
}
__host__ __device__ __forceinline__ constexpr int PIDX(int a, int b) {
  return (a > b) ? IDX(a, b) : IDX(b, a);
}

__global__ __launch_bounds__(BLOCK_T, 1) void spodnet_kernel(
    const float* __restrict__ theta, const float* __restrict__ nz,
    float* __restrict__ out) {
  __shared__ float lds[BLOCK_T * MAT_F];   // 51200 B
  __shared__ float snz[90];                // staged noise (10 x 9)
  const int tid = threadIdx.x;
  const size_t blockBase = (size_t)blockIdx.x * (BLOCK_T * MAT_F);
  const int mb = tid * MAT_F;              // this lane's LDS slot base (index)

  // ---------------- stage input: global -> LDS (per-wave TDM) --------------
#if HAVE_TDM
  {
    const int wl = __builtin_amdgcn_readfirstlane((int)tid); // uniform
    const int wave = wl >> 5;
    const unsigned lbyte =
        (unsigned)(size_t)(const void*)&lds[wave * DW_PER_WAVE];
    const unsigned long long ga =
        (unsigned long long)(size_t)(theta + blockBase +
                                     (size_t)wave * DW_PER_WAVE);
    tdm_copy_1d(true, lbyte, ga, (unsigned)DW_PER_WAVE);
  }
#else
  for (int k = tid; k < BLOCK_T * MAT_F; k += BLOCK_T)
    lds[k] = theta[blockBase + k];
#endif

  // Stage noise into LDS while the TDM DMA runs.
  if (tid < 90) snz[tid] = nz[tid];
  __syncthreads();           // publishes snz; TDM slice is wave-local
  wait_tensorcnt0();         // each wave waits for its own tensor DMA
  sched_fence();

  // =================== PHASE A: W = inv(Theta) -> LDS ======================
  {
    // Read lower triangle of Theta (input slot is consumed here).
    float th[55];
#pragma unroll
    for (int i = 0; i < 10; ++i)
#pragma unroll
      for (int j = 0; j <= i; ++j) th[IDX(i, j)] = lds[mb + i * 10 + j];
    sched_fence();

    // Cholesky (diagonal of th holds 1/L[k][k]).
#pragma unroll
    for (int k = 0; k < 10; ++k) {
      float d = th[IDX(k, k)];
#pragma unroll
      for (int m = 0; m < k; ++m) d -= th[IDX(k, m)] * th[IDX(k, m)];
      const float si = fast_rsqrt(d);
      th[IDX(k, k)] = si;
#pragma unroll
      for (int i = k + 1; i < 10; ++i) {
        float t = th[IDX(i, k)];
#pragma unroll
        for (int m = 0; m < k; ++m) t -= th[IDX(i, m)] * th[IDX(k, m)];
        th[IDX(i, k)] = t * si;
      }
    }
    sched_fence();

    // mi = L^{-1}; th dies here.
    float mi[55];
#pragma unroll
    for (int j = 0; j < 10; ++j) {
      mi[IDX(j, j)] = th[IDX(j, j)];                     // = 1/L[j][j]
#pragma unroll
      for (int i = j + 1; i < 10; ++i) {
        float acc = th[IDX(i, j)] * mi[IDX(j, j)];       // m = j term
#pragma unroll
        for (int m = j + 1; m < i; ++m) acc += th[IDX(i, m)] * mi[IDX(m, j)];
        mi[IDX(i, j)] = -acc * th[IDX(i, i)];
      }
    }
    sched_fence();

    // lauum: W[i][j] = sum_{k>=i} mi[k][i]*mi[k][j], streamed straight to the
    // lane's LDS slot (packed, indices 0..54) -- no W register array here.
#pragma unroll
    for (int i = 0; i < 10; ++i)
#pragma unroll
      for (int j = 0; j <= i; ++j) {
        float acc = 0.0f;
#pragma unroll
        for (int k = i; k < 10; ++k) acc += mi[IDX(k, i)] * mi[IDX(k, j)];
        lds[mb + IDX(i, j)] = acc;
      }
  }
  sched_fence();

  // =================== PHASE B: 10 column passes ===========================
  // Reload packed W (contiguous -> ds_load_b128s). Fresh register web.
  float w[55];
#pragma unroll
  for (int t = 0; t < 55; ++t) w[t] = lds[mb + t];
  sched_fence();

#pragma unroll
  for (int col = 0; col < 10; ++col) {
    float v[9], w12[9];
#pragma unroll
    for (int j = 0; j < 9; ++j) v[j] = snz[col * 9 + j];  // LDS broadcast
    const float w22 = w[IDX(col, col)];
    const float rw = fast_rcp(w22);
#pragma unroll
    for (int i = 0; i < 9; ++i) {
      const int ri = (i < col) ? i : i + 1;
      w12[i] = w[PIDX(ri, col)];
    }
    float u[9];  // u = W11 * v ; later reused as w12n
#pragma unroll
    for (int i = 0; i < 9; ++i) {
      const int ri = (i < col) ? i : i + 1;
      float acc = 0.0f;
#pragma unroll
      for (int j = 0; j < 9; ++j) {
        const int rj = (j < col) ? j : j + 1;
        acc += w[PIDX(ri, rj)] * v[j];
      }
      u[i] = acc;
    }
    float schur = 0.0f, c = 0.0f;
#pragma unroll
    for (int i = 0; i < 9; ++i) {
      schur += u[i] * v[i];
      c += w12[i] * v[i];
    }
    lds[mb + col * 10 + col] = GY + schur; // t22n -> output diagonal (ds);
                                           // clobbers stale packed W: fine,
                                           // W is register-resident now.
    const float cr = c * rw;
    // t22n - v' inv_T11 v == GY + c^2/w22  (algebraic simplification)
    const float w22n = fast_rcp(GY + c * cr);
    // u <- w12n = -w22n * (u - cr*w12)   (z = inv_T11 * v, in place)
#pragma unroll
    for (int i = 0; i < 9; ++i) u[i] = -w22n * (u[i] - cr * w12[i]);
    // W11 <- W11 - rw*w12 w12' + w22n*u u'   (two symmetric rank-1s)
#pragma unroll
    for (int i = 0; i < 9; ++i) {
      const int ri = (i < col) ? i : i + 1;
      const float pi = rw * w12[i];
      const float qi = w22n * u[i];
#pragma unroll
      for (int j = i; j < 9; ++j) {
        const int rj = (j < col) ? j : j + 1;
        w[PIDX(ri, rj)] += qi * u[j] - pi * w12[j];
      }
    }
#pragma unroll
    for (int i = 0; i < 9; ++i) {
      const int ri = (i < col) ? i : i + 1;
      w[PIDX(ri, col)] = u[i];
    }
    w[IDX(col, col)] = w22n;
  }
  sched_fence();

  // ---------------- batch-independent off-diagonal outputs -----------------
  // Theta_out[i][j] = noise[max(i,j)*9 + min(i,j)]; all 100 output positions
  // get written (90 here + 10 diagonals above).
#pragma unroll
  for (int i = 0; i < 10; ++i)
#pragma unroll
    for (int j = 0; j < 10; ++j) {
      if (i == j) continue;
      const int mx = (i > j) ? i : j;
      const int mn = (i > j) ? j : i;
      lds[mb + i * 10 + j] = snz[mx * 9 + mn];
    }

  __syncthreads();  // commit ds_stores before the TDM store reads LDS

  // ---------------- stream output: LDS -> global (per-wave TDM) ------------
#if HAVE_TDM
  {
    const int wl = __builtin_amdgcn_readfirstlane((int)tid);
    const int wave = wl >> 5;
    const unsigned lbyte =
        (unsigned)(size_t)(const void*)&lds[wave * DW_PER_WAVE];
    const unsigned long long ga =
        (unsigned long long)(size_t)(out + blockBase +
                                     (size_t)wave * DW_PER_WAVE);
    tdm_copy_1d(false, lbyte, ga, (unsigned)DW_PER_WAVE);
    wait_tensorcnt0();
  }
#else
  for (int k = tid; k < BLOCK_T * MAT_F; k += BLOCK_T)
    out[blockBase + k] = lds[k];
#endif
}

extern "C" void kernel_launch(void* const* d_in, const int* in_sizes, int n_in,
                              void* d_out, int out_size, void* d_ws,
                              size_t ws_size, hipStream_t stream) {
  const float* theta = (const float*)d_in[0];  // (B,10,10) f32
  const float* noise = (const float*)d_in[1];  // (1,10,9)  f32
  float* out = (float*)d_out;                  // (B,10,10) f32
  const int nmat = in_sizes[0] / MAT_F;        // 262144
  const int blocks = nmat / BLOCK_T;           // 2048
  spodnet_kernel<<<blocks, BLOCK_T, 0, stream>>>(theta, noise, out);
  (void)n_in; (void)out_size; (void)d_ws; (void)ws_size;
}